// ScalableAlgebraicTransformer_8615704396099
// MI455X (gfx1250) — compile-verified
//
#include <hip/hip_runtime.h>
#include <hip/hip_bf16.h>
#include <math.h>

// Problem constants (match reference)
#define Bn     1024
#define Sn     256
#define Gn     128
#define Dn     32
#define DEPTHn 3
#define En     8192   // R*D*D
#define BLKn   1024   // D*D

typedef __attribute__((ext_vector_type(2))) float v2f;
typedef __attribute__((ext_vector_type(8))) float v8f;
typedef __attribute__((ext_vector_type(4))) int   v4i;

__device__ __forceinline__ float nan0(float v) {
  if (__builtin_isnan(v)) return 0.0f;
  if (__builtin_isinf(v)) return v > 0.0f ? 3.402823466e38f : -3.402823466e38f;
  return v;
}

__device__ __forceinline__ float waveSum(float v) {
  #pragma unroll
  for (int m = 16; m >= 1; m >>= 1) v += __shfl_xor(v, m, 32);
  return v;
}

// D = A(16x4,f32) * B(4x16,f32) + C(16x16,f32), wave32 WMMA
__device__ __forceinline__ v8f wmma_f32(v2f a, v2f b, v8f c) {
  return __builtin_amdgcn_wmma_f32_16x16x4_f32(false, a, false, b, (short)0, c,
                                               false, false);
}

// ---- CDNA5 async global->LDS copy (guarded; sync fallback keeps compile safe)
#if __has_builtin(__builtin_amdgcn_global_load_async_to_lds_b128)
#define USE_ASYNC_LDS 1
typedef __attribute__((address_space(1))) v4i g_v4i;  // global b128
typedef __attribute__((address_space(3))) v4i l_v4i;  // LDS b128
__device__ __forceinline__ void async_copy_b128(const float* g, float* l) {
  // flat shared pointer: low 32 bits are the LDS byte offset on AMDGPU
  __builtin_amdgcn_global_load_async_to_lds_b128(
      (g_v4i*)(unsigned long long)(uintptr_t)g,
      (l_v4i*)(unsigned)(uintptr_t)l, 0, 0);
}
__device__ __forceinline__ void wait_async0() {
#if __has_builtin(__builtin_amdgcn_s_wait_asynccnt)
  __builtin_amdgcn_s_wait_asynccnt(0);
#else
  asm volatile("s_wait_asynccnt 0x0" ::: "memory");
#endif
}
#else
#define USE_ASYNC_LDS 0
#endif

// ---------------------------------------------------------------------------
// Kernel 1: reps_n = nan_to_num(reps) / (frobenius_norm + 1e-6), per (g,r)
// ---------------------------------------------------------------------------
__global__ void reps_norm_kernel(const float* __restrict__ reps,
                                 float* __restrict__ reps_n) {
  __shared__ float wsum[8];
  const int mat = blockIdx.x;     // g*R + r
  const int tid = threadIdx.x;    // 0..255, one float4 each
  const float4* src = (const float4*)(reps + (size_t)mat * BLKn);
  float4 v = src[tid];
  v.x = nan0(v.x); v.y = nan0(v.y); v.z = nan0(v.z); v.w = nan0(v.w);
  float s = v.x*v.x + v.y*v.y + v.z*v.z + v.w*v.w;
  s = waveSum(s);
  if ((tid & 31) == 0) wsum[tid >> 5] = s;
  __syncthreads();
  if (tid == 0) {
    float t = 0.f;
    #pragma unroll
    for (int i = 0; i < 8; ++i) t += wsum[i];
    wsum[0] = 1.0f / (sqrtf(t) + 1e-6f);
  }
  __syncthreads();
  const float sc = wsum[0];
  float4 o; o.x = v.x*sc; o.y = v.y*sc; o.z = v.z*sc; o.w = v.w*sc;
  ((float4*)(reps_n + (size_t)mat * BLKn))[tid] = o;
}

// ---------------------------------------------------------------------------
// Kernel 2: sequential scan. One workgroup (8 waves) per batch element;
// wave r owns block r. x in LDS row-major; M in LDS in PAIR-INTERLEAVED
// layout ms[r][k/2][col][k&1] so WMMA B fragments are single ds_load_b64.
// Next step's M is register-prefetched (pair-rows) overlapping compute.
// ---------------------------------------------------------------------------
__global__ void scan_kernel(const int* __restrict__ seq,
                            const float* __restrict__ reps,
                            const float* __restrict__ reps_n,
                            const int* __restrict__ trans,
                            const float* __restrict__ lng,
                            const float* __restrict__ lnb,
                            const float* __restrict__ og,
                            const float* __restrict__ ob,
                            const float* __restrict__ b2,
                            float* __restrict__ xn,
                            float* __restrict__ out) {
  extern __shared__ float smem[];
  float* xs   = smem;            // En: state x, row-major per block
  float* ms   = smem + En;       // En: M, pair-swizzled per block
  float* redS = smem + 2 * En;   // 8
  float* redQ = redS + 8;        // 8
  float* stat = redQ + 8;        // 2 (mu, rstd)

  const int b    = blockIdx.x;
  const int tid  = threadIdx.x;
  const int w    = tid >> 5;      // wave id == block index r
  const int lane = tid & 31;
  const int half = lane >> 4;
  const int ln   = lane & 15;

  // ---- x0 = nan_to_num(reps[seq[b,0]]) (UNnormalized reps) ----
  {
    const int g0 = seq[b * Sn + 0];
    const float4* src = (const float4*)(reps + (size_t)g0 * En);
    float4* dst = (float4*)xs;
    #pragma unroll
    for (int j = 0; j < En / 4 / 256; ++j) {
      float4 v = src[tid + j * 256];
      v.x = nan0(v.x); v.y = nan0(v.y); v.z = nan0(v.z); v.w = nan0(v.w);
      dst[tid + j * 256] = v;
    }
  }

  // ---- prefetch M for step 0: pair-rows (2k, 2k+1) x 4 cols per unit ----
  float4 plo[4], phi[4];
  {
    const int t = trans[seq[b * Sn + 0]];
    const float* base = reps_n + (size_t)t * En;
    #pragma unroll
    for (int j = 0; j < 4; ++j) {
      const int u = tid + j * 256;                  // 0..1023 units
      const int r = u >> 7, kp = (u >> 3) & 15, c4 = u & 7;
      const float* p = base + r * BLKn + (2 * kp) * Dn + 4 * c4;
      plo[j] = *(const float4*)p;
      phi[j] = *(const float4*)(p + Dn);
    }
  }

  const float* Xr = xs + w * BLKn;
  const float* Mw = ms + w * BLKn;

  for (int s = 0; s < Sn; ++s) {
    __syncthreads();                   // prev step done with ms / x0 ready
    // store prefetched M, pair-interleaved: ms[r*1024 + kp*64 + col*2 + par]
    #pragma unroll
    for (int j = 0; j < 4; ++j) {
      const int u = tid + j * 256;
      const int r = u >> 7, kp = (u >> 3) & 15, c4 = u & 7;
      float* d = ms + r * BLKn + kp * 64 + (4 * c4) * 2;
      v2f t0 = {plo[j].x, phi[j].x}; *(v2f*)(d + 0) = t0;
      v2f t1 = {plo[j].y, phi[j].y}; *(v2f*)(d + 2) = t1;
      v2f t2 = {plo[j].z, phi[j].z}; *(v2f*)(d + 4) = t2;
      v2f t3 = {plo[j].w, phi[j].w}; *(v2f*)(d + 6) = t3;
    }
    __syncthreads();
    if (s + 1 < Sn) {                  // prefetch next step's M (overlaps compute)
      const int t = trans[seq[b * Sn + s + 1]];
      const float* base = reps_n + (size_t)t * En;
      #pragma unroll
      for (int j = 0; j < 4; ++j) {
        const int u = tid + j * 256;
        const int r = u >> 7, kp = (u >> 3) & 15, c4 = u & 7;
        const float* p = base + r * BLKn + (2 * kp) * Dn + 4 * c4;
        plo[j] = *(const float4*)p;
        phi[j] = *(const float4*)(p + Dn);
      }
    }

    #pragma unroll
    for (int l = 0; l < DEPTHn; ++l) {
      // C = X_r @ M_r, 32x32x32 as 2x2 tiles x 8 K-chunks. Per k0 only 4 frag
      // registers live -> no spills. All loads are aligned b64.
      v8f c[2][2] = {};
      #pragma unroll
      for (int k0 = 0; k0 < 8; ++k0) {
        const int kb = 4 * k0 + 2 * half;
        v2f a0 = *(const v2f*)(Xr + ln * Dn + kb);
        v2f a1 = *(const v2f*)(Xr + (16 + ln) * Dn + kb);
        const float* bp_ = Mw + (2 * k0 + half) * 64 + 2 * ln;
        v2f b0 = *(const v2f*)bp_;
        v2f b1 = *(const v2f*)(bp_ + 32);
        c[0][0] = wmma_f32(a0, b0, c[0][0]);
        c[0][1] = wmma_f32(a0, b1, c[0][1]);
        c[1][0] = wmma_f32(a1, b0, c[1][0]);
        c[1][1] = wmma_f32(a1, b1, c[1][1]);
      }

      // residual add + moments (C layout: row = v + 8*half, col = ln)
      float lsum = 0.f, lsq = 0.f;
      #pragma unroll
      for (int i = 0; i < 2; ++i)
        #pragma unroll
        for (int j = 0; j < 2; ++j)
          #pragma unroll
          for (int v = 0; v < 8; ++v) {
            const int row = 16 * i + v + 8 * half;
            const int col = 16 * j + ln;
            const float u = c[i][j][v] + Xr[row * Dn + col];
            c[i][j][v] = u;
            lsum += u; lsq += u * u;
          }
      lsum = waveSum(lsum); lsq = waveSum(lsq);
      if (lane == 0) { redS[w] = lsum; redQ[w] = lsq; }
      __syncthreads();
      if (tid == 0) {
        float tS = 0.f, tQ = 0.f;
        #pragma unroll
        for (int i = 0; i < 8; ++i) { tS += redS[i]; tQ += redQ[i]; }
        const float mu  = tS * (1.0f / En);
        const float var = tQ * (1.0f / En) - mu * mu;   // biased variance
        stat[0] = mu; stat[1] = rsqrtf(var + 1e-3f);
      }
      __syncthreads();
      const float mu = stat[0], rstd = stat[1];
      const float* gp = lng + l * En;
      const float* bp = lnb + l * En;
      float* Xw = xs + w * BLKn;
      #pragma unroll
      for (int i = 0; i < 2; ++i)
        #pragma unroll
        for (int j = 0; j < 2; ++j)
          #pragma unroll
          for (int v = 0; v < 8; ++v) {
            const int row = 16 * i + v + 8 * half;
            const int col = 16 * j + ln;
            const int e   = w * BLKn + row * Dn + col;
            Xw[row * Dn + col] = (c[i][j][v] - mu) * rstd * gp[e] + bp[e];
          }
    }
  }

  // ---- output layernorm (eps 1e-5) -> xn; out[b] initialized with b2 ----
  __syncthreads();
  float lsum = 0.f, lsq = 0.f;
  const int base = tid * 32;
  #pragma unroll
  for (int k = 0; k < 32; ++k) { const float u = xs[base + k]; lsum += u; lsq += u * u; }
  lsum = waveSum(lsum); lsq = waveSum(lsq);
  if (lane == 0) { redS[w] = lsum; redQ[w] = lsq; }
  __syncthreads();
  if (tid == 0) {
    float tS = 0.f, tQ = 0.f;
    #pragma unroll
    for (int i = 0; i < 8; ++i) { tS += redS[i]; tQ += redQ[i]; }
    const float mu  = tS * (1.0f / En);
    const float var = tQ * (1.0f / En) - mu * mu;
    stat[0] = mu; stat[1] = rsqrtf(var + 1e-5f);
  }
  __syncthreads();
  const float mu = stat[0], rstd = stat[1];
  float* xrow = xn + (size_t)b * En;
  #pragma unroll
  for (int k = 0; k < 32; ++k) {
    const int e = base + k;
    xrow[e] = (xs[e] - mu) * rstd * og[e] + ob[e];
  }
  if (tid == 0) out[b] = b2[0];
}

// ---------------------------------------------------------------------------
// Kernel 3: out += relu(xn @ W1 + b1) @ W2, fused; double-buffered K loop.
// A tile (64x32, row-major) staged with GLOBAL_LOAD_ASYNC_TO_LDS_B128 when
// available (reg fallback otherwise). B tile (32x64) staged via registers and
// stored pair-interleaved Bs[kp*128 + nn*2 + par] -> b64 WMMA B fragments.
// ---------------------------------------------------------------------------
__global__ void mlp_kernel(const float* __restrict__ xn,
                           const float* __restrict__ W1,
                           const float* __restrict__ b1,
                           const float* __restrict__ W2,
                           float* __restrict__ out) {
  __shared__ __align__(16) float As[2][64 * 32];
  __shared__ __align__(16) float Bs[2][16 * 128];
  const int tid  = threadIdx.x;
  const int w    = tid >> 5;
  const int lane = tid & 31;
  const int half = lane >> 4;
  const int ln   = lane & 15;
  const int i    = w >> 1;             // row tile 0..3
  const int j0   = (w & 1) * 2;        // first of two col tiles
  const int rb   = blockIdx.y * 64;
  const int cb   = blockIdx.x * 64;
  const int kpB  = tid >> 4;           // B staging: pair-row 0..15
  const int c4B  = tid & 15;           // B staging: 4-col group 0..15

  v8f c0 = {}; v8f c1 = {};
  float4 blo, bhi;
#if !USE_ASYNC_LDS
  float4 areg[2];
#endif

  // ---- prologue: stage chunk kc=0 into buffer 0 ----
  #pragma unroll
  for (int t = 0; t < 2; ++t) {
    const int idx = tid + t * 256;
    const int rr = idx >> 3, c4 = idx & 7;
#if USE_ASYNC_LDS
    async_copy_b128(xn + (size_t)(rb + rr) * En + 4 * c4, &As[0][rr * 32 + 4 * c4]);
#else
    *(float4*)&As[0][rr * 32 + 4 * c4] =
        *(const float4*)(xn + (size_t)(rb + rr) * En + 4 * c4);
#endif
  }
  {
    const float* p = W1 + (size_t)(2 * kpB) * En + cb + 4 * c4B;
    blo = *(const float4*)p; bhi = *(const float4*)(p + En);
    float* d = &Bs[0][kpB * 128 + (4 * c4B) * 2];
    v2f t0 = {blo.x, bhi.x}; *(v2f*)(d + 0) = t0;
    v2f t1 = {blo.y, bhi.y}; *(v2f*)(d + 2) = t1;
    v2f t2 = {blo.z, bhi.z}; *(v2f*)(d + 4) = t2;
    v2f t3 = {blo.w, bhi.w}; *(v2f*)(d + 6) = t3;
  }

  for (int kc = 0; kc < En; kc += 32) {
    const int cur = (kc >> 5) & 1, nxt = cur ^ 1;
    const bool more = (kc + 32) < En;
#if USE_ASYNC_LDS
    wait_async0();                      // own async writes to LDS landed
#endif
    __syncthreads();                    // all waves past buffers; [cur] ready
    if (more) {                         // issue next-chunk loads (hidden by compute)
      #pragma unroll
      for (int t = 0; t < 2; ++t) {
        const int idx = tid + t * 256;
        const int rr = idx >> 3, c4 = idx & 7;
#if USE_ASYNC_LDS
        async_copy_b128(xn + (size_t)(rb + rr) * En + kc + 32 + 4 * c4,
                        &As[nxt][rr * 32 + 4 * c4]);
#else
        areg[t] = *(const float4*)(xn + (size_t)(rb + rr) * En + kc + 32 + 4 * c4);
#endif
      }
      const float* p = W1 + (size_t)(kc + 32 + 2 * kpB) * En + cb + 4 * c4B;
      blo = *(const float4*)p; bhi = *(const float4*)(p + En);
    }

    const float* Ac = As[cur];
    const float* Bc = Bs[cur];
    #pragma unroll
    for (int k0 = 0; k0 < 8; ++k0) {
      const int kb = 4 * k0 + 2 * half;
      v2f a = *(const v2f*)(Ac + (16 * i + ln) * 32 + kb);
      const float* bp_ = Bc + (2 * k0 + half) * 128 + 2 * (16 * j0 + ln);
      v2f bA = *(const v2f*)bp_;
      v2f bB = *(const v2f*)(bp_ + 32);
      c0 = wmma_f32(a, bA, c0);
      c1 = wmma_f32(a, bB, c1);
    }

    if (more) {                         // store next B (and A fallback) to [nxt]
#if !USE_ASYNC_LDS
      #pragma unroll
      for (int t = 0; t < 2; ++t) {
        const int idx = tid + t * 256;
        const int rr = idx >> 3, c4 = idx & 7;
        *(float4*)&As[nxt][rr * 32 + 4 * c4] = areg[t];
      }
#endif
      float* d = &Bs[nxt][kpB * 128 + (4 * c4B) * 2];
      v2f t0 = {blo.x, bhi.x}; *(v2f*)(d + 0) = t0;
      v2f t1 = {blo.y, bhi.y}; *(v2f*)(d + 2) = t1;
      v2f t2 = {blo.z, bhi.z}; *(v2f*)(d + 4) = t2;
      v2f t3 = {blo.w, bhi.w}; *(v2f*)(d + 6) = t3;
    }
  }

  // ---- epilogue: bias + relu + dot W2, per-row shfl sum, atomic accumulate
  const int col0 = cb + 16 * j0 + ln;
  const int col1 = cb + 16 * (j0 + 1) + ln;
  const float bb0 = b1[col0], bb1 = b1[col1];
  const float w20 = W2[col0], w21 = W2[col1];
  #pragma unroll
  for (int v = 0; v < 8; ++v) {
    float h0 = c0[v] + bb0; h0 = h0 > 0.f ? h0 : 0.f;
    float h1 = c1[v] + bb1; h1 = h1 > 0.f ? h1 : 0.f;
    float t = h0 * w20 + h1 * w21;
    #pragma unroll
    for (int m = 1; m <= 8; m <<= 1) t += __shfl_xor(t, m, 32);  // per 16-lane half
    if (ln == 0) {
      const int row = rb + 16 * i + v + 8 * half;
      atomicAdd(&out[row], t);
    }
  }
}

// ---------------------------------------------------------------------------
extern "C" void kernel_launch(void* const* d_in, const int* in_sizes, int n_in,
                              void* d_out, int out_size, void* d_ws, size_t ws_size,
                              hipStream_t stream) {
  (void)in_sizes; (void)n_in; (void)out_size; (void)ws_size;
  const int*   seq   = (const int*)  d_in[0];   // [B,S]
  const float* reps  = (const float*)d_in[1];   // [G,R,D,D]
  const int*   trans = (const int*)  d_in[2];   // [G]
  const float* lng   = (const float*)d_in[3];   // [DEPTH,E]
  const float* lnb   = (const float*)d_in[4];   // [DEPTH,E]
  const float* og    = (const float*)d_in[5];   // [E]
  const float* ob    = (const float*)d_in[6];   // [E]
  const float* W1    = (const float*)d_in[7];   // [E,E]
  const float* b1    = (const float*)d_in[8];   // [E]
  const float* W2    = (const float*)d_in[9];   // [E,1]
  const float* b2    = (const float*)d_in[10];  // [1]
  float* out = (float*)d_out;                   // [B]

  float* reps_n = (float*)d_ws;                              // 4 MB
  float* xn     = reps_n + (size_t)Gn * 8 * Dn * Dn;         // 32 MB

  reps_norm_kernel<<<Gn * 8, 256, 0, stream>>>(reps, reps_n);

  const size_t smem = (size_t)(2 * En + 32) * sizeof(float); // 65,664 B
  scan_kernel<<<Bn, 256, smem, stream>>>(seq, reps, reps_n, trans,
                                         lng, lnb, og, ob, b2, xn, out);

  dim3 gB(En / 64, Bn / 64);  // (128, 16)
  mlp_kernel<<<gB, 256, 0, stream>>>(xn, W1, b1, W2, out);
}